// MaskedPooling_63359357551135
// MI455X (gfx1250) — compile-verified
//
#include <hip/hip_runtime.h>

typedef __attribute__((ext_vector_type(2))) float v2f;
typedef __attribute__((ext_vector_type(8))) float v8f;

#define B_SZ  64
#define K_SEG 19
#define C_CH  384
#define HW    256     // 16*16 flattened
#define RD    256     // region_dim
#define M_PAD 32      // 20 rows padded to 2 WMMA M-tiles

// ---------------------------------------------------------------------------
// Kernel 1: build Aw [B, 32, 256]:
//   rows 0..18 : bilinear-downsampled seg weights * (1/256)
//   row  19    : 1/256 (global-mean row, fused as a matmul row)
//   rows 20..31: 0 (padding so WMMA M-tiles are clean)
// Half-pixel bilinear at scale 16: out(i,j) = 0.25 * sum of input pixels
// (16i+7..16i+8, 16j+7..16j+8). Only 1/64 of seg is ever read.
// ---------------------------------------------------------------------------
__global__ void k_downsample(const float* __restrict__ seg, float* __restrict__ wA) {
    int idx = blockIdx.x * blockDim.x + threadIdx.x;   // B*32*256 threads
    int b   = idx >> 13;          // / (32*256)
    int rem = idx & 8191;
    int k   = rem >> 8;           // 0..31
    int hw  = rem & 255;
    int i   = hw >> 4, j = hw & 15;
    float v = 0.0f;
    if (k < K_SEG) {
        const float* p = seg + (((size_t)(b * K_SEG + k)) << 16);  // 256*256 plane
        const float* q = p + (i * 16 + 7) * 256 + (j * 16 + 7);
        v = 0.25f * (q[0] + q[1] + q[256] + q[257]) * (1.0f / 256.0f);
    } else if (k == K_SEG) {
        v = 1.0f / 256.0f;
    }
    wA[idx] = v;
}

// ---------------------------------------------------------------------------
// Kernel 2: feats[b, 0..31, 0..383] = Aw[b] (32x256) x F[b]^T (256x384)
// One wave per 16x16 tile; K stepped by 4 via V_WMMA_F32_16X16X4_F32.
// Fragment layout (ISA 7.12.2): lane = half*16 + mn
//   A: a = A[mn, kk+2*half .. +1]           (float2)
//   B: b = F[c=mn_col, kk+2*half .. +1]     (B[k,n] = F^T => row-major F load)
//   D: vgpr r -> row r + 8*half
// 8 waves/block, 6 blocks/batch cover 2 Mtiles x 24 Ctiles = 48 tiles.
// ---------------------------------------------------------------------------
__global__ void k_gemm_region(const float* __restrict__ wA, const float* __restrict__ Fm,
                              float* __restrict__ feats) {
    int b    = blockIdx.x / 6;
    int t    = (blockIdx.x % 6) * 8 + (threadIdx.x >> 5);   // tile id 0..47
    int lane = threadIdx.x & 31;
    int half = lane >> 4, mn = lane & 15;
    int mtile = t & 1, ctile = t >> 1;                      // 2 x 24 tiles

    const float* A  = wA + b * (M_PAD * HW) + (mtile * 16 + mn) * HW + 2 * half;
    const float* Bp = Fm + (size_t)b * (C_CH * HW) + (ctile * 16 + mn) * HW + 2 * half;

    v8f acc = {0.f, 0.f, 0.f, 0.f, 0.f, 0.f, 0.f, 0.f};
    for (int kk = 0; kk < HW; kk += 4) {
        v2f av = *(const v2f*)(A + kk);
        v2f bv = *(const v2f*)(Bp + kk);
        acc = __builtin_amdgcn_wmma_f32_16x16x4_f32(false, av, false, bv,
                                                    (short)0, acc, false, false);
    }

    float* out = feats + b * (M_PAD * C_CH) + (mtile * 16 + 8 * half) * C_CH
               + ctile * 16 + mn;
#pragma unroll
    for (int r = 0; r < 8; ++r) out[r * C_CH] = acc[r];
}

// ---------------------------------------------------------------------------
// Kernel 3: out[b, m<20, d] = relu( feats[b] (32x384) x W^T (384x256) + bias )
// Same tiling: 2 Mtiles x 16 Dtiles = 32 tiles, 8 waves/block, 4 blocks/batch.
// WMMA runs fully convergent; the row<20 predicate only guards stores.
// ---------------------------------------------------------------------------
__global__ void k_gemm_proj(const float* __restrict__ feats, const float* __restrict__ W,
                            const float* __restrict__ bias, float* __restrict__ out) {
    int b    = blockIdx.x / 4;
    int t    = (blockIdx.x % 4) * 8 + (threadIdx.x >> 5);   // tile id 0..31
    int lane = threadIdx.x & 31;
    int half = lane >> 4, mn = lane & 15;
    int mtile = t & 1, dtile = t >> 1;                      // 2 x 16 tiles

    const float* A  = feats + b * (M_PAD * C_CH) + (mtile * 16 + mn) * C_CH + 2 * half;
    const float* Bp = W + (dtile * 16 + mn) * C_CH + 2 * half;

    v8f acc = {0.f, 0.f, 0.f, 0.f, 0.f, 0.f, 0.f, 0.f};
    for (int kk = 0; kk < C_CH; kk += 4) {
        v2f av = *(const v2f*)(A + kk);
        v2f bv = *(const v2f*)(Bp + kk);
        acc = __builtin_amdgcn_wmma_f32_16x16x4_f32(false, av, false, bv,
                                                    (short)0, acc, false, false);
    }

    float bb = bias[dtile * 16 + mn];
#pragma unroll
    for (int r = 0; r < 8; ++r) {
        int row = mtile * 16 + 8 * half + r;
        if (row < 20) {
            out[(size_t)b * ((K_SEG + 1) * RD) + row * RD + dtile * 16 + mn] =
                fmaxf(acc[r] + bb, 0.0f);
        }
    }
}

// ---------------------------------------------------------------------------
extern "C" void kernel_launch(void* const* d_in, const int* in_sizes, int n_in,
                              void* d_out, int out_size, void* d_ws, size_t ws_size,
                              hipStream_t stream) {
    const float* F_map = (const float*)d_in[0];   // [64,384,16,16]
    const float* seg   = (const float*)d_in[1];   // [64,19,256,256]
    const float* W     = (const float*)d_in[2];   // [256,384]
    const float* bias  = (const float*)d_in[3];   // [256]
    float*       out   = (float*)d_out;           // [64, 20*256]

    float* wA    = (float*)d_ws;                       // B*32*256 f32 = 2 MB
    float* feats = wA + (size_t)B_SZ * M_PAD * HW;     // B*32*384 f32 = 3 MB

    // 1) downsample + build padded A matrix (one thread per element)
    k_downsample<<<B_SZ * M_PAD, 256, 0, stream>>>(seg, wA);

    // 2) region/global features: 48 WMMA tiles per batch, 8 waves per block
    k_gemm_region<<<B_SZ * 6, 256, 0, stream>>>(wA, F_map, feats);

    // 3) projection + bias + ReLU: 32 WMMA tiles per batch
    k_gemm_proj<<<B_SZ * 4, 256, 0, stream>>>(feats, W, bias, out);
}